// WeightSharedNegativeSampling_28810640621864
// MI455X (gfx1250) — compile-verified
//
#include <hip/hip_runtime.h>

// Reference: word2vec-style negative sampling.
//   h          (B,D)   f32
//   emb_table  (V,D)   f32   (V=1e6 -> 512MB, random row gathers)
//   target_idx (B,)    int
//   neg_idx    (B,S)   int
// out = [pos_out(B), pos_label(B)=1, neg_out(B*S), neg_label(B*S)=0]  (f32)
//
// Gather-bound (~60MB traffic, ~2.6us floor @23.3TB/s). Dot products done via
// f32 WMMA (V_WMMA_F32_16X16X4_F32) to keep f32 precision on the matrix path.

#define B_SIZE   16384
#define D_MODEL  128
#define S_NEG    5
#define WAVES_PER_BLOCK 8
#define ROWS_PER_WAVE   16
#define KSTEPS   (D_MODEL / 4)   // 32 chained K=4 WMMAs

typedef __attribute__((ext_vector_type(2))) float v2f;
typedef __attribute__((ext_vector_type(8))) float v8f;

__global__ __launch_bounds__(32 * WAVES_PER_BLOCK)
void WeightSharedNegativeSampling_kernel(
    const float* __restrict__ h,
    const float* __restrict__ emb,
    const int*   __restrict__ tgt,
    const int*   __restrict__ neg,
    float*       __restrict__ out)
{
    __shared__ float stage[WAVES_PER_BLOCK][16 * 16];

    const int lane = threadIdx.x & 31;
    const int wave = threadIdx.x >> 5;
    const int tile = blockIdx.x * WAVES_PER_BLOCK + wave;
    const int base = tile * ROWS_PER_WAVE;        // first batch row of this tile

    // Both A (16x4, M in lanes) and B (4x16, N in lanes) fragments for f32 WMMA
    // reduce to: lane L holds float2 at row[L&15]*D + 4k + 2*(L>=16).
    const int mrow = lane & 15;
    const int koff = (lane >> 4) << 1;            // 0 or 2

    // --- Preload A fragments (h tile), reused for all 6 passes -------------
    const float* hrow = h + (size_t)(base + mrow) * D_MODEL + koff;
    v2f a[KSTEPS];
#pragma unroll
    for (int k = 0; k < KSTEPS; ++k)
        a[k] = *(const v2f*)(hrow + k * 4);

    float* pos_out = out;                                   // B
    float* pos_lab = out + B_SIZE;                          // B
    float* neg_out = out + 2 * B_SIZE;                      // B*S
    float* neg_lab = out + 2 * B_SIZE + B_SIZE * S_NEG;     // B*S

    // --- 6 passes: j=0 positive, j=1..5 negatives --------------------------
    for (int j = 0; j < 1 + S_NEG; ++j) {
        // Embedding row feeding column n = mrow of the B fragment.
        const int ridx = (j == 0) ? tgt[base + mrow]
                                  : neg[(base + mrow) * S_NEG + (j - 1)];
        const float* erow = emb + (size_t)ridx * D_MODEL + koff;

        v8f acc = {};
#pragma unroll
        for (int k = 0; k < KSTEPS; ++k) {
            v2f b = *(const v2f*)(erow + k * 4);
            // D = A*B + C ; 8 args: (neg_a, A, neg_b, B, c_mod, C, reuse_a, reuse_b)
            acc = __builtin_amdgcn_wmma_f32_16x16x4_f32(
                false, a[k], false, b, (short)0, acc, false, false);
        }

        // D layout: VGPR r, lanes 0-15 -> (M=r, N=lane); lanes 16-31 -> (M=r+8).
        // Stage 16x16 tile to LDS, then lanes 0..15 pull the diagonal.
        float* t = &stage[wave][0];
#pragma unroll
        for (int r = 0; r < 8; ++r)
            t[(r + ((lane >> 4) << 3)) * 16 + (lane & 15)] = acc[r];
        // Same-wave LDS ops are in-order; load below gets its own dscnt wait.
        if (lane < 16) {
            float d   = t[lane * 16 + lane];          // h_row . e_row
            float sig = 1.0f / (1.0f + __expf(-d));   // sigmoid
            int   row = base + lane;
            if (j == 0) pos_out[row] = sig;
            else        neg_out[row * S_NEG + (j - 1)] = sig;
        }
    }

    // --- Labels ------------------------------------------------------------
    if (lane < 16) pos_lab[base + lane] = 1.0f;
#pragma unroll
    for (int it = 0; it < 3; ++it) {
        int i = it * 32 + lane;
        if (i < ROWS_PER_WAVE * S_NEG)
            neg_lab[base * S_NEG + i] = 0.0f;
    }
}

extern "C" void kernel_launch(void* const* d_in, const int* in_sizes, int n_in,
                              void* d_out, int out_size, void* d_ws, size_t ws_size,
                              hipStream_t stream) {
    const float* h   = (const float*)d_in[0];
    const float* emb = (const float*)d_in[1];
    const int*   tgt = (const int*)d_in[2];
    const int*   neg = (const int*)d_in[3];
    float*       out = (float*)d_out;

    dim3 grid(B_SIZE / (ROWS_PER_WAVE * WAVES_PER_BLOCK));   // 128 blocks
    dim3 block(32 * WAVES_PER_BLOCK);                        // 8 waves (wave32)
    hipLaunchKernelGGL(WeightSharedNegativeSampling_kernel, grid, block, 0, stream,
                       h, emb, tgt, neg, out);
}